// SPUTransformer_62002147885333
// MI455X (gfx1250) — compile-verified
//
#include <hip/hip_runtime.h>

// Native clang vector type: required for __builtin_nontemporal_{load,store},
// and guarantees global_load_b128 / global_store_b128 codegen.
typedef float v4f __attribute__((ext_vector_type(4)));

#define SPU_LOG2E 1.44269504088896340736f

// spu(x) = x>=0 ? x*x - 0.5 : sigmoid(-x) - 1
// sigmoid(-x) - 1 = 1/(1+e^x) - 1 = -e^x / (1 + e^x)
__device__ __forceinline__ float spu_f(float x) {
    float q = __builtin_fmaf(x, x, -0.5f);                 // pos branch
    float e = __builtin_amdgcn_exp2f(x * SPU_LOG2E);       // v_exp_f32
    float s = -e * __builtin_amdgcn_rcpf(1.0f + e);        // v_rcp_f32 (neg branch)
    return x >= 0.0f ? q : s;                              // branchless select
}

struct LoHi { float lo, hi; };

// One row of the DeepPoly-style SPU relaxation (matches the JAX reference).
__device__ __forceinline__ LoHi spu_relax(float l, float u,
                                          float p1l, float p1u,
                                          float q1l, float q1u,
                                          float b0l, float b0u) {
    float sl = spu_f(l);
    float su = spu_f(u);

    bool neg   = (u <= 0.0f);
    bool pos   = (l >= 0.0f);
    bool cross = !(neg || pos);

    // w >= 0.01 guarantees u - l >= 0.02, so the divide is safe.
    float all_slopes = (su - sl) / (u - l);

    float s0 = neg ? all_slopes : 0.0f;   // s0 = where(neg, slopes, 0)
    float s1 = neg ? 0.0f : all_slopes;   // pos|cross == !neg

    bool neg_slope = all_slopes < 0.0f;
    float lo = neg_slope ? su : sl;
    float hi = neg_slope ? sl : su;
    lo = cross ? -0.5f : lo;

    float sh1 = __builtin_fmaf(-s1, u, su);   // su - s1*u
    float sh0 = __builtin_fmaf(-s0, l, sl);   // sl - s0*l
    sh0 = cross ? -0.5f : sh0;
    sh1 = neg   ? sl    : sh1;

    float s1p = fmaxf(s1, 0.0f), s1n = fminf(s1, 0.0f);
    float s0p = fmaxf(s0, 0.0f), s0n = fminf(s0, 0.0f);

    float UBM = __builtin_fmaf(s1p, p1u, s1n * p1l);
    float UBV = __builtin_fmaf(s1p, q1u, __builtin_fmaf(s1n, q1l, sh1));
    float LBM = __builtin_fmaf(s0p, p1l, s0n * p1u);
    float LBV = __builtin_fmaf(s0n, q1l, __builtin_fmaf(s0p, q1u, sh0));

    float lower = __builtin_fmaf(fmaxf(LBM, 0.0f), b0l,
                  __builtin_fmaf(fminf(LBM, 0.0f), b0u, LBV));
    float upper = __builtin_fmaf(fmaxf(UBM, 0.0f), b0u,
                  __builtin_fmaf(fminf(UBM, 0.0f), b0l, UBV));

    lo = (lower > lo) ? lower : lo;   // where(lower > lo, lower, lo)
    hi = (upper < hi) ? upper : hi;   // where(upper < hi, upper, hi)

    LoHi r;
    r.lo = lo;
    r.hi = hi;
    return r;
}

// Each thread processes GROUPS_PER_THREAD float4 groups (2 rows each),
// block-strided so every B128 access stays fully coalesced across the wave32.
#define TPB 256
#define GROUPS_PER_THREAD 2

__global__ __launch_bounds__(TPB) void spu_transformer_kernel(
    const v4f* __restrict__ bounds,
    const v4f* __restrict__ slopes_prev,
    const v4f* __restrict__ shifts_prev,
    const v4f* __restrict__ bounds_prev,
    v4f* __restrict__ out,
    int n4)   // number of float4 groups = (2*N)/4
{
    int base = blockIdx.x * (TPB * GROUPS_PER_THREAD) + threadIdx.x;
#pragma unroll
    for (int k = 0; k < GROUPS_PER_THREAD; ++k) {
        int i = base + k * TPB;
        if (i < n4) {
            // Streaming: non-temporal B128 loads/stores (th:*_NT on gfx1250).
            v4f b  = __builtin_nontemporal_load(&bounds[i]);
            v4f sp = __builtin_nontemporal_load(&slopes_prev[i]);
            v4f sh = __builtin_nontemporal_load(&shifts_prev[i]);
            v4f bp = __builtin_nontemporal_load(&bounds_prev[i]);

            LoHi r0 = spu_relax(b.x, b.y, sp.x, sp.y, sh.x, sh.y, bp.x, bp.y);
            LoHi r1 = spu_relax(b.z, b.w, sp.z, sp.w, sh.z, sh.w, bp.z, bp.w);

            v4f r;
            r.x = r0.lo;
            r.y = r0.hi;
            r.z = r1.lo;
            r.w = r1.hi;
            __builtin_nontemporal_store(r, &out[i]);
        }
    }
}

extern "C" void kernel_launch(void* const* d_in, const int* in_sizes, int n_in,
                              void* d_out, int out_size, void* d_ws, size_t ws_size,
                              hipStream_t stream) {
    const v4f* bounds      = (const v4f*)d_in[0];
    const v4f* slopes_prev = (const v4f*)d_in[1];
    const v4f* shifts_prev = (const v4f*)d_in[2];
    const v4f* bounds_prev = (const v4f*)d_in[3];
    v4f* out = (v4f*)d_out;

    int n_flat = in_sizes[0];     // 2*N floats (N = 8388608 -> divisible by 4)
    int n4 = n_flat / 4;          // float4 groups

    int per_block = TPB * GROUPS_PER_THREAD;
    int grid = (n4 + per_block - 1) / per_block;

    spu_transformer_kernel<<<grid, TPB, 0, stream>>>(
        bounds, slopes_prev, shifts_prev, bounds_prev, out, n4);
}